// MLPPredictor_48009144435110
// MI455X (gfx1250) — compile-verified
//
#include <hip/hip_runtime.h>
#include <hip/hip_bf16.h>

typedef __attribute__((ext_vector_type(2))) float v2f;
typedef __attribute__((ext_vector_type(4))) float v4f;
typedef __attribute__((ext_vector_type(8))) float v8f;

#define D_IN 128
#define OUTW 8

// ---------------------------------------------------------------------------
// Phase 1: P[n][0:8] = h[n] . Wu^T ; P[n][8:16] = h[n] . Wv^T
// One wave32 computes a 16-node tile via 32 steps of V_WMMA_F32_16X16X4_F32.
// Wcat[n][k] = (n<8) ? W[n][k] : W[n-8][128+k]   (W is 8x256 row-major)
// ---------------------------------------------------------------------------
__global__ __launch_bounds__(256) void node_proj_kernel(
    const float* __restrict__ h, const float* __restrict__ W,
    float* __restrict__ P, int n_nodes) {
  const int lane = threadIdx.x & 31;
  const int wave = threadIdx.x >> 5;
  const int m0 = (blockIdx.x * 8 + wave) * 16;   // wave-uniform tile base
  if (m0 >= n_nodes) return;                     // uniform: EXEC stays all-1s

  const int half = lane >> 4;    // 0: K%4 in {0,1}; 1: K%4 in {2,3}
  const int lm   = lane & 15;

  // B fragment source: B[k][n] = Wcat[n][k]; this lane supplies n = lm.
  const float* wbase = (lm < 8) ? (W + lm * 256) : (W + (lm - 8) * 256 + 128);

  // Preload all 32 B fragments (Wcat is tiny; stays in VGPRs).
  v2f bfrag[32];
#pragma unroll
  for (int k = 0; k < 32; ++k)
    bfrag[k] = *(const v2f*)(wbase + 4 * k + 2 * half);

  // A fragment source: row m0+lm of h, K offset 2*half within each step.
  int arow = m0 + lm;
  if (arow >= n_nodes) arow = n_nodes - 1;       // safety clamp (no-op here)
  const float* abase = h + (size_t)arow * D_IN + 2 * half;

  v8f acc = {0.f, 0.f, 0.f, 0.f, 0.f, 0.f, 0.f, 0.f};
#pragma unroll
  for (int k = 0; k < 32; ++k) {
    v2f afrag = *(const v2f*)(abase + 4 * k);
    acc = __builtin_amdgcn_wmma_f32_16x16x4_f32(
        /*neg_a=*/false, afrag, /*neg_b=*/false, bfrag[k],
        /*c_mod=*/(short)0, acc, /*reuse_a=*/false, /*reuse_b=*/false);
  }

  // D layout: vgpr j holds P[m0 + j + 8*half][lm] -> coalesced b32 stores.
  float* prow = P + (size_t)(m0 + 8 * half) * 16 + lm;
#pragma unroll
  for (int j = 0; j < 8; ++j) {
    int row = m0 + 8 * half + j;
    if (row < n_nodes) prow[(size_t)j * 16] = acc[j];
  }
}

// ---------------------------------------------------------------------------
// Phase 2: out[e][o] = P[src[e]][o] + P[dst[e]][8+o] + b[o]
// One thread per edge; 2x v4f gathers from L2-resident P, NT v4f stores.
// ---------------------------------------------------------------------------
__global__ __launch_bounds__(256) void edge_score_kernel(
    const float* __restrict__ P, const int* __restrict__ src,
    const int* __restrict__ dst, const float* __restrict__ b,
    float* __restrict__ out, int n_edges) {
  int e = blockIdx.x * blockDim.x + threadIdx.x;
  if (e >= n_edges) return;

  const v4f* pu = (const v4f*)(P + (size_t)src[e] * 16);
  const v4f* pv = (const v4f*)(P + (size_t)dst[e] * 16 + 8);
  const v4f* bb = (const v4f*)b;

  v4f u0 = pu[0], u1 = pu[1];
  v4f v0 = pv[0], v1 = pv[1];
  v4f b0 = bb[0], b1 = bb[1];

  v4f r0 = u0 + v0 + b0;
  v4f r1 = u1 + v1 + b1;

  v4f* o = (v4f*)(out + (size_t)e * OUTW);
  __builtin_nontemporal_store(r0, o);       // write-once 51 MB stream:
  __builtin_nontemporal_store(r1, o + 1);   // keep P hot in near caches
}

extern "C" void kernel_launch(void* const* d_in, const int* in_sizes, int n_in,
                              void* d_out, int out_size, void* d_ws, size_t ws_size,
                              hipStream_t stream) {
  const float* h   = (const float*)d_in[0];   // (N_NODES, 128) fp32
  const int*   src = (const int*)d_in[1];     // (N_EDGES,)
  const int*   dst = (const int*)d_in[2];     // (N_EDGES,)
  const float* W   = (const float*)d_in[3];   // (8, 256) fp32
  const float* b   = (const float*)d_in[4];   // (8,) fp32
  float* out = (float*)d_out;                 // (N_EDGES, 8) fp32

  const int n_nodes = in_sizes[0] / D_IN;
  const int n_edges = in_sizes[1];

  float* P = (float*)d_ws;                    // (n_nodes, 16) fp32 = 6.4 MB

  const int tiles = (n_nodes + 15) / 16;
  const int blocks1 = (tiles + 7) / 8;        // 8 waves (tiles) per 256-thr block
  node_proj_kernel<<<blocks1, 256, 0, stream>>>(h, W, P, n_nodes);

  const int blocks2 = (n_edges + 255) / 256;
  edge_score_kernel<<<blocks2, 256, 0, stream>>>(P, src, dst, b, out, n_edges);
}